// cheb_conv_kagcn_5377299055094
// MI455X (gfx1250) — compile-verified
//
#include <hip/hip_runtime.h>
#include <hip/hip_bf16.h>
#include <math.h>

// Problem constants
#define BB 64
#define NN 34
#define NP 48   // padded N (3 wmma tiles)
#define FF 16
#define TT 96
#define KK_CHEB 3
#define OO 64

typedef __attribute__((ext_vector_type(2))) float v2f;
typedef __attribute__((ext_vector_type(8))) float v8f;

__device__ __forceinline__ v8f wmma4(v2f a, v2f b, v8f c) {
    // V_WMMA_F32_16X16X4_F32 : 8-arg pattern (neg_a, A, neg_b, B, c_mod, C, reuse_a, reuse_b)
    return __builtin_amdgcn_wmma_f32_16x16x4_f32(false, a, false, b, (short)0, c, false, false);
}

// C[m0..m0+15, n0..n0+15] = A(MxK, lda) @ B(KxN, ldb), K = KK (multiple of 4)
__device__ __forceinline__ void gemm_tile(const float* Ab, int lda,
                                          const float* Bb, int ldb,
                                          float* Cb, int ldc,
                                          int m0, int n0, int KK, int lane) {
    const int half = lane >> 4;
    const int l    = lane & 15;
    v8f acc = {};
#pragma unroll
    for (int k0 = 0; k0 < 48; k0 += 4) {
        if (k0 >= KK) break;
        v2f a, b;
        a.x = Ab[(m0 + l) * lda + k0 + 2 * half];
        a.y = Ab[(m0 + l) * lda + k0 + 2 * half + 1];
        b.x = Bb[(k0 + 2 * half) * ldb + n0 + l];
        b.y = Bb[(k0 + 2 * half + 1) * ldb + n0 + l];
        acc = wmma4(a, b, acc);
    }
#pragma unroll
    for (int r = 0; r < 8; ++r)
        Cb[(m0 + half * 8 + r) * ldc + n0 + l] = acc[r];
}

// C[m0.., n0c..] = A^T @ B where A is stored K-major (K x M, lda), B is (K x 16, ldb)
__device__ __forceinline__ void gemm_tile_ta(const float* Ab, int lda,
                                             const float* Bb, int ldb,
                                             float* Cb, int ldc,
                                             int m0, int n0c, int KK, int lane) {
    const int half = lane >> 4;
    const int l    = lane & 15;
    v8f acc = {};
#pragma unroll
    for (int k0 = 0; k0 < 48; k0 += 4) {
        if (k0 >= KK) break;
        v2f a, b;
        a.x = Ab[(k0 + 2 * half) * lda + m0 + l];
        a.y = Ab[(k0 + 2 * half + 1) * lda + m0 + l];
        b.x = Bb[(k0 + 2 * half) * ldb + l];
        b.y = Bb[(k0 + 2 * half + 1) * ldb + l];
        acc = wmma4(a, b, acc);
    }
#pragma unroll
    for (int r = 0; r < 8; ++r)
        Cb[(m0 + half * 8 + r) * ldc + n0c + l] = acc[r];
}

// OUT tile = relu(YY(48x96) @ TH(96x64)) stored transposed to out[b][j][o][t]
__device__ __forceinline__ void gemm_out_tile(const float* Yb, const float* Tb,
                                              float* __restrict__ outg,
                                              int b, int t, int m0, int n0, int lane) {
    const int half = lane >> 4;
    const int l    = lane & 15;
    v8f acc = {};
#pragma unroll
    for (int k0 = 0; k0 < 96; k0 += 4) {
        v2f a, bv;
        a.x  = Yb[(m0 + l) * 96 + k0 + 2 * half];
        a.y  = Yb[(m0 + l) * 96 + k0 + 2 * half + 1];
        bv.x = Tb[(k0 + 2 * half) * OO + n0 + l];
        bv.y = Tb[(k0 + 2 * half + 1) * OO + n0 + l];
        acc = wmma4(a, bv, acc);
    }
#pragma unroll
    for (int r = 0; r < 8; ++r) {
        int j = m0 + half * 8 + r;
        if (j < NN) {
            int o = n0 + l;
            float v = acc[r];
            v = v > 0.f ? v : 0.f;
            outg[(((size_t)b * NN + j) * OO + o) * TT + t] = v;
        }
    }
}

// -------- kernel 1: softmax of st_attention over axis 2 (the i index) --------
__global__ void __launch_bounds__(64)
kagcn_softmax(const float* __restrict__ st, float* __restrict__ att) {
    const int bk = blockIdx.x;            // b*K + k
    const int j  = threadIdx.x;
    if (j >= NN) return;
    const float* p = st + (size_t)bk * NN * NN;
    float*       q = att + (size_t)bk * NN * NN;
    float mx = -1e30f;
    for (int i = 0; i < NN; ++i) mx = fmaxf(mx, p[i * NN + j]);
    float s = 0.f;
    for (int i = 0; i < NN; ++i) s += __expf(p[i * NN + j] - mx);
    float inv = 1.f / s;
    for (int i = 0; i < NN; ++i) q[i * NN + j] = __expf(p[i * NN + j] - mx) * inv;
}

// -------- kernel 2: per-(b,t) full pipeline --------
__global__ void __launch_bounds__(128)
kagcn_main(const float* __restrict__ x, const float* __restrict__ att,
           const float* __restrict__ position, const float* __restrict__ distance,
           const float* __restrict__ Theta, const float* __restrict__ ThetaL,
           float* __restrict__ out) {
    extern __shared__ float sm[];
    float* sGS  = sm;            // 48*16   = 768
    float* sTH  = sm + 768;      // 96*64   = 6144
    float* sA   = sm + 6912;     // 48*48   = 2304
    float* sL1  = sm + 9216;     // 2304
    float* sL2  = sm + 11520;    // 2304
    float* sQ1  = sm + 13824;    // 2304
    float* sQ2  = sm + 16128;    // 2304
    float* sMk  = sm + 18432;    // 2304
    float* sYY  = sm + 20736;    // 48*96   = 4608
    float* svn  = sm + 25344;    // 48
    float* spn  = sm + 25392;    // 48
    float* sdi  = sm + 25440;    // 48
    float* sdc  = sm + 25488;    // 48
    float* sRed = sm + 25536;    // 128
    float* sScal= sm + 25664;    // 2

    const int tid  = threadIdx.x;
    const int lane = tid & 31;
    const int wave = tid >> 5;
    const int bid  = blockIdx.x;
    const int b    = bid / TT;
    const int t    = bid % TT;

    // ---- stage 0: load gs tile (zero-padded) and stacked [Theta;Theta_L] ----
    for (int idx = tid; idx < NP * FF; idx += 128) {
        int n = idx >> 4, f = idx & 15;
        sGS[idx] = (n < NN) ? x[(((size_t)b * NN + n) * FF + f) * TT + t] : 0.f;
    }
    for (int idx = tid; idx < 96 * OO; idx += 128) {
        int r = idx >> 6, o = idx & 63;
        int g = r >> 4, f = r & 15;
        int k = g >> 1;
        const float* src = (g & 1) ? ThetaL : Theta;
        sTH[idx] = src[((size_t)k * FF + f) * OO + o];
    }
    __syncthreads();

    // ---- stage 1: vector norms ----
    if (tid < NP) {
        float v1 = (tid < NN) ? sGS[tid * FF + 1] : 0.f;
        float v2 = (tid < NN) ? sGS[tid * FF + 2] : 0.f;
        svn[tid] = sqrtf(v1 * v1 + v2 * v2);
        float p0 = 0.f, p1 = 0.f;
        if (tid < NN) { p0 = position[tid * 2]; p1 = position[tid * 2 + 1]; }
        spn[tid] = sqrtf(p0 * p0 + p1 * p1);
    }
    __syncthreads();

    // ---- stage 2: dynamic adjacency + Frobenius norm ----
    float fsum = 0.f;
    for (int idx = tid; idx < NP * NP; idx += 128) {
        int n = idx / NP, m = idx % NP;
        float a = 0.f;
        if (n < NN && m < NN) {
            float v1 = sGS[n * FF + 1], v2 = sGS[n * FF + 2];
            float dot = v1 * position[m * 2] + v2 * position[m * 2 + 1];
            float c = dot / (fmaxf(svn[n], 1e-8f) * fmaxf(spn[m], 1e-8f));
            c = fmaxf(c, 0.f);
            a = c * svn[n] / distance[n * NN + m];
        }
        sA[idx] = a;
        fsum += a * a;
    }
    sRed[tid] = fsum;
    __syncthreads();
    if (tid == 0) {
        float s = 0.f;
        for (int i = 0; i < 128; ++i) s += sRed[i];
        sScal[0] = 1.f / fmaxf(sqrtf(s), 1e-12f);
    }
    __syncthreads();
    float rinv = sScal[0];
    for (int idx = tid; idx < NP * NP; idx += 128) sA[idx] *= rinv;
    __syncthreads();

    // ---- row / column degree inverses ----
    if (tid < 2 * NP) {
        int isCol = tid >= NP;
        int q = isCol ? tid - NP : tid;
        float d = 0.f;
        if (isCol) { for (int n = 0; n < NP; ++n) d += sA[n * NP + q]; }
        else       { for (int m = 0; m < NP; ++m) d += sA[q * NP + m]; }
        float di = (d > 0.f) ? rsqrtf(fmaxf(d, 1e-12f)) : 0.f;
        if (isCol) sdc[q] = di; else sdi[q] = di;
    }
    __syncthreads();

    // ---- stage 3: L1 = lap(A), L2 = lap(A^T) ----
    for (int idx = tid; idx < NP * NP; idx += 128) {
        int n = idx / NP, m = idx % NP;
        float eye = (n == m && n < NN) ? 1.f : 0.f;
        sL1[idx] = eye - sdi[n] * sA[idx] * sdi[m];
        sL2[idx] = eye - sdc[n] * sA[m * NP + n] * sdc[m];
    }
    __syncthreads();

    // ---- stage 4: Q = L @ L (Chebyshev T2 ingredient), 18 wmma tiles ----
    for (int tIdx = wave; tIdx < 18; tIdx += 4) {
        int matId = tIdx / 9, p = tIdx % 9;
        const float* Lp = matId ? sL2 : sL1;
        float*       Qp = matId ? sQ2 : sQ1;
        gemm_tile(Lp, NP, Lp, NP, Qp, NP, (p / 3) * 16, (p % 3) * 16, NP, lane);
    }
    __syncthreads();

    // ---- stage 5: for each (k, side): Mk = att_k (.) Tk ; Y panel = Mk^T @ gs ----
    for (int g = 0; g < 6; ++g) {
        int k = g >> 1, side = g & 1;
        const float* base = att + ((size_t)b * KK_CHEB + k) * NN * NN;
        for (int idx = tid; idx < NP * NP; idx += 128) {
            int i = idx / NP, j = idx % NP;
            float mval = 0.f;
            if (i < NN && j < NN) {
                float av = base[i * NN + j];
                float tk;
                if (k == 0)      tk = (i == j) ? 1.f : 0.f;
                else if (k == 1) tk = (side ? sL2 : sL1)[idx];
                else             tk = 2.f * (side ? sQ2 : sQ1)[idx] - ((i == j) ? 1.f : 0.f);
                mval = av * tk;
            }
            sMk[idx] = mval;
        }
        __syncthreads();
        if (wave < 3)
            gemm_tile_ta(sMk, NP, sGS, FF, sYY, 96, wave * 16, g * 16, NP, lane);
        __syncthreads();
    }

    // ---- stage 7: OUT = relu(YY @ TH), 12 tiles, fused transposed store ----
    for (int p = wave; p < 12; p += 4)
        gemm_out_tile(sYY, sTH, out, b, t, (p / 4) * 16, (p % 4) * 16, lane);
}

extern "C" void kernel_launch(void* const* d_in, const int* in_sizes, int n_in,
                              void* d_out, int out_size, void* d_ws, size_t ws_size,
                              hipStream_t stream) {
    const float* x    = (const float*)d_in[0];
    const float* st   = (const float*)d_in[1];
    const float* pos  = (const float*)d_in[2];
    const float* dist = (const float*)d_in[3];
    const float* Th   = (const float*)d_in[4];
    const float* ThL  = (const float*)d_in[5];
    float* out = (float*)d_out;
    float* att = (float*)d_ws;   // B*K*N*N floats = ~888 KB

    kagcn_softmax<<<BB * KK_CHEB, 64, 0, stream>>>(st, att);

    size_t shmem = 25666 * sizeof(float);   // ~100 KB dynamic LDS (320 KB/WGP available)
    kagcn_main<<<BB * TT, 128, shmem, stream>>>(x, att, pos, dist, Th, ThL, out);
}